// DRQN_net_55791625175403
// MI455X (gfx1250) — compile-verified
//
#include <hip/hip_runtime.h>
#include <hip/hip_bf16.h>

// MI455X / gfx1250, wave32. One wave per GRU chain (4096 waves total).
// Front-end GEMMs batched 16 timesteps at a time via V_WMMA_F32_16X16X4_F32.

typedef float v2f __attribute__((ext_vector_type(2)));
typedef float v8f __attribute__((ext_vector_type(8)));

#define WAVES_PER_BLOCK 4
#define CHUNK 16

__device__ __forceinline__ float rdlanef(float v, int lane) {
  return __builtin_bit_cast(float,
      __builtin_amdgcn_readlane(__builtin_bit_cast(int, v), lane));
}

#if __has_builtin(__builtin_amdgcn_exp2f)
#define FAST_EXP2(x) __builtin_amdgcn_exp2f(x)
#else
#define FAST_EXP2(x) exp2f(x)
#endif
#if __has_builtin(__builtin_amdgcn_rcpf)
#define FAST_RCP(x) __builtin_amdgcn_rcpf(x)
#else
#define FAST_RCP(x) (1.0f / (x))
#endif

__device__ __forceinline__ float fast_tanh(float x) {
#if __has_builtin(__builtin_amdgcn_tanhf)
  return __builtin_amdgcn_tanhf(x);          // v_tanh_f32: 1 trans op
#else
  float e = FAST_EXP2(-2.885390082f * x);    // exp(-2x)
  return (1.0f - e) * FAST_RCP(1.0f + e);
#endif
}
__device__ __forceinline__ float fast_sigmoid(float x) {
  return 0.5f * fast_tanh(0.5f * x) + 0.5f;
}

__global__ __launch_bounds__(32 * WAVES_PER_BLOCK)
void drqn_gru_fused(const float* __restrict__ x, const float* __restrict__ hx,
                    const float* __restrict__ fc1_w, const float* __restrict__ fc1_b,
                    const float* __restrict__ w_ih, const float* __restrict__ w_hh,
                    const float* __restrict__ b_ih, const float* __restrict__ b_hh,
                    const float* __restrict__ fc2_w, const float* __restrict__ fc2_b,
                    float* __restrict__ out, int B, int S)
{
  __shared__ float lds[WAVES_PER_BLOCK][CHUNK * 4];   // 16x4 feats tile / wave

  const int lane  = threadIdx.x & 31;
  const int wv    = threadIdx.x >> 5;
  const int b     = blockIdx.x * WAVES_PER_BLOCK + wv;   // wave-uniform
  if (b >= B) return;

  const int n     = lane & 15;            // column / row-in-tile index
  const int khalf = (lane >> 4) << 1;     // 0 or 2 : K pair held by this lane
  const int hi8   = (lane >> 4) << 3;     // 0 or 8 : D rows M=j / M=j+8

  // ---- Loop-invariant WMMA B-operands (4x16, zero-padded) --------------
  // B1[k][n] = fc1_w^T[k][n] = fc1_w[n*4+k];  B2[k][n] = w_ih^T[k][n]
  v2f b1, b2;
  b1.x = (n < 4) ? fc1_w[n * 4 + khalf]     : 0.0f;
  b1.y = (n < 4) ? fc1_w[n * 4 + khalf + 1] : 0.0f;
  b2.x = (n < 3) ? w_ih[n * 4 + khalf]      : 0.0f;
  b2.y = (n < 3) ? w_ih[n * 4 + khalf + 1]  : 0.0f;

  const float bias1 = (n < 4) ? fc1_b[n] : 0.0f;   // per-column fc1 bias
  const float bias2 = (n < 3) ? b_ih[n]  : 0.0f;   // per-column gate bias

  // GRU / fc2 scalars (H == 1)
  const float whh_r = w_hh[0], whh_z = w_hh[1], whh_n = w_hh[2];
  const float bhh_r = b_hh[0], bhh_z = b_hh[1], bhh_n = b_hh[2];
  const float fw = fc2_w[lane & 1], fb = fc2_b[lane & 1];
  const int pair = lane >> 1;             // lane -> (step, action) for y store

  float h = hx[b];
  const float* __restrict__ xb = x + (size_t)b * S * 4;
  float* __restrict__ yb = out + (size_t)b * S * 2;
  float* __restrict__ l0 = &lds[wv][0];

  for (int s0 = 0; s0 < S; s0 += CHUNK) {
    if (s0 + CHUNK < S)   // global_prefetch_b8 of next chunk's A-tile
      __builtin_prefetch(xb + (size_t)(s0 + CHUNK + n) * 4 + khalf, 0, 1);

    // ---- A-tile: 16 timesteps x 4 features, ISA 16x4 A-layout ----------
    v2f a;
    { const float* p = xb + (size_t)(s0 + n) * 4 + khalf;
      a.x = p[0]; a.y = p[1]; }

    v8f c0 = {};
    // feats_raw(16x16, cols 0..3 live) = A(16x4) x B1(4x16)
    v8f d1 = __builtin_amdgcn_wmma_f32_16x16x4_f32(
        false, a, false, b1, (short)0, c0, false, false);

    // bias + ReLU, transpose D-layout -> A-layout via per-wave LDS tile
    if (n < 4) {
#pragma unroll
      for (int j = 0; j < 8; ++j)
        l0[(j + hi8) * 4 + n] = fmaxf(d1[j] + bias1, 0.0f);
    }
    asm volatile("s_wait_dscnt 0" ::: "memory");   // same-wave DS RAW fence
    v2f a2;
    { const float* p = l0 + n * 4 + khalf;
      a2.x = p[0]; a2.y = p[1]; }

    // gates(16x16, cols 0..2 live) = feats(16x4) x B2(4x16)
    v8f d2 = __builtin_amdgcn_wmma_f32_16x16x4_f32(
        false, a2, false, b2, (short)0, c0, false, false);
#pragma unroll
    for (int j = 0; j < 8; ++j) d2[j] += bias2;    // per-column b_ih

    // ---- 16 serial GRU steps, wave-uniform via readlane ----------------
    float hv = 0.0f;                               // per-lane h for y store
#pragma unroll
    for (int i = 0; i < CHUNK; ++i) {
      const int j  = i & 7;                        // D VGPR holding row i
      const int lb = (i < 8) ? 0 : 16;             // lane base for row i
      const float gr = rdlanef(d2[j], lb + 0);
      const float gz = rdlanef(d2[j], lb + 1);
      const float gn = rdlanef(d2[j], lb + 2);
      const float r  = fast_sigmoid(gr + h * whh_r + bhh_r);
      const float z  = fast_sigmoid(gz + h * whh_z + bhh_z);
      const float nn = fast_tanh(gn + r * (h * whh_n + bhh_n));
      h = (1.0f - z) * nn + z * h;
      hv = (pair == i) ? h : hv;                   // fan h_i out to lanes 2i,2i+1
    }

    // ---- coalesced y store: 32 floats = 16 steps x 2 actions -----------
    yb[(size_t)s0 * 2 + lane] = hv * fw + fb;
  }

  if (lane == 0) out[(size_t)B * S * 2 + b] = h;   // hT, flat after y
}

extern "C" void kernel_launch(void* const* d_in, const int* in_sizes, int n_in,
                              void* d_out, int out_size, void* d_ws, size_t ws_size,
                              hipStream_t stream) {
  const float* x     = (const float*)d_in[0];
  const float* hx    = (const float*)d_in[1];
  const float* fc1_w = (const float*)d_in[2];
  const float* fc1_b = (const float*)d_in[3];
  const float* w_ih  = (const float*)d_in[4];
  const float* w_hh  = (const float*)d_in[5];
  const float* b_ih  = (const float*)d_in[6];
  const float* b_hh  = (const float*)d_in[7];
  const float* fc2_w = (const float*)d_in[8];
  const float* fc2_b = (const float*)d_in[9];
  float* out = (float*)d_out;

  const int B = in_sizes[1];                 // hx: (1,B,1)
  const int S = in_sizes[0] / (B * 4);       // x:  (B,S,4)

  dim3 block(32 * WAVES_PER_BLOCK);
  dim3 grid((B + WAVES_PER_BLOCK - 1) / WAVES_PER_BLOCK);
  hipLaunchKernelGGL(drqn_gru_fused, grid, block, 0, stream,
                     x, hx, fc1_w, fc1_b, w_ih, w_hh, b_ih, b_hh,
                     fc2_w, fc2_b, out, B, S);
}